// FP4Quantizer_43490838839436
// MI455X (gfx1250) — compile-verified
//
#include <hip/hip_runtime.h>
#include <math.h>

// ---------------------------------------------------------------------------
// FP4 block quant-dequant (straight-through forward) for MI455X / gfx1250.
// Pass 1: stage x into LDS via async global->LDS (CDNA5 path), per-block
//         top-5 |x| selection -> 0.95-quantile scale; WG-reduce min/max of
//         scales into global accumulators (uint-bit atomics, scales > 0).
// Pass 2: double-quantize scale with global min/max, snap x to FP4 levels
//         via scaled midpoint thresholds, write x + (x_q - x).
// ---------------------------------------------------------------------------

#define TPB1 128          // threads per WG in pass 1; 1 thread == 1 block of 64
#define PAD  68           // LDS words per 64-float block (16B aligned, ~2-way)

#if __has_builtin(__builtin_amdgcn_global_load_async_to_lds_b128)
#define HAVE_ASYNC_LDS 1
typedef int v4i __attribute__((vector_size(16)));                 // matches builtin param
typedef __attribute__((address_space(1))) v4i GV4;                // global (__device__)
typedef __attribute__((address_space(3))) v4i LV4;                // LDS (__shared__)
#else
#define HAVE_ASYNC_LDS 0
#endif

__global__ void fp4_init_mm(unsigned* __restrict__ mm) {
  mm[0] = 0x7F800000u;   // +inf bits  (min accumulator; positive-float order == uint order)
  mm[1] = 0x00000000u;   // 0.0f bits  (max accumulator)
}

__device__ __forceinline__ void top5_insert(float a, float& t0, float& t1,
                                            float& t2, float& t3, float& t4) {
  float b;
  b = fminf(a, t0); t0 = fmaxf(a, t0); a = b;
  b = fminf(a, t1); t1 = fmaxf(a, t1); a = b;
  b = fminf(a, t2); t2 = fmaxf(a, t2); a = b;
  b = fminf(a, t3); t3 = fmaxf(a, t3);
  t4 = fmaxf(b, t4);
}

__global__ __launch_bounds__(TPB1)
void fp4_pass1_scales(const float* __restrict__ x,
                      float* __restrict__ scales,
                      unsigned* __restrict__ mm,
                      int nblocks) {
  __shared__ float stage[TPB1 * PAD];       // 128 blocks * 68 words = 34 KB
  __shared__ float rmin[TPB1], rmax[TPB1];

  const int tid = threadIdx.x;
  const long long wgChunk0 = (long long)blockIdx.x * (TPB1 * 16);  // float4 units
  const float4* xg4 = (const float4*)x;

  // ---- stage 128 blocks (32 KB) into LDS, fully coalesced 16B/lane ----
#pragma unroll
  for (int i = 0; i < 16; ++i) {
    const int c   = i * TPB1 + tid;       // local float4 chunk id
    const int blk = c >> 4;               // local block (64 floats = 16 chunks)
    const int e   = (c & 15) << 2;        // word offset inside block
    float* dst = &stage[blk * PAD + e];
    const float4* src = xg4 + (wgChunk0 + c);
#if HAVE_ASYNC_LDS
    __builtin_amdgcn_global_load_async_to_lds_b128((GV4*)src, (LV4*)dst, 0, 0);
#else
    *reinterpret_cast<float4*>(dst) = *src;
#endif
  }
#if HAVE_ASYNC_LDS
#if __has_builtin(__builtin_amdgcn_s_wait_asynccnt)
  __builtin_amdgcn_s_wait_asynccnt(0);
#else
  asm volatile("s_wait_asynccnt 0" ::: "memory");
#endif
#endif
  __syncthreads();

  // ---- per-thread: top-5 |x| of my 64-element block ----
  float t0 = 0.f, t1 = 0.f, t2 = 0.f, t3 = 0.f, t4 = 0.f;
#pragma unroll
  for (int j = 0; j < 16; ++j) {
    const float4 v = *reinterpret_cast<const float4*>(&stage[tid * PAD + 4 * j]);
    top5_insert(fabsf(v.x), t0, t1, t2, t3, t4);
    top5_insert(fabsf(v.y), t0, t1, t2, t3, t4);
    top5_insert(fabsf(v.z), t0, t1, t2, t3, t4);
    top5_insert(fabsf(v.w), t0, t1, t2, t3, t4);
  }

  // quantile(|x|, 0.95) over 64 values, linear interpolation:
  // index = 0.95*(64-1); v[59] = 5th largest = t4, v[60] = 4th largest = t3
  const float idxf = 0.95f * 63.0f;
  float scale = t4 * (60.0f - idxf) + t3 * (idxf - 59.0f);
  scale = fmaxf(scale, 1e-8f);

  const int gblk = blockIdx.x * TPB1 + tid;
  if (gblk < nblocks) scales[gblk] = scale;

  // ---- WG reduce min/max of scales, one atomic pair per WG ----
  rmin[tid] = scale;
  rmax[tid] = scale;
  __syncthreads();
#pragma unroll
  for (int s = TPB1 / 2; s > 0; s >>= 1) {
    if (tid < s) {
      rmin[tid] = fminf(rmin[tid], rmin[tid + s]);
      rmax[tid] = fmaxf(rmax[tid], rmax[tid + s]);
    }
    __syncthreads();
  }
  if (tid == 0) {
    atomicMin(&mm[0], __float_as_uint(rmin[0]));   // scales > 0: uint order ok
    atomicMax(&mm[1], __float_as_uint(rmax[0]));
  }
}

__global__ __launch_bounds__(256)
void fp4_pass2_quantdq(const float* __restrict__ x,
                       const float* __restrict__ scales,
                       const unsigned* __restrict__ mm,
                       float* __restrict__ out,
                       long long n) {
  const long long gid = (long long)blockIdx.x * 256 + threadIdx.x;  // float4 idx
  const long long e0  = gid * 4;
  if (e0 >= n) return;

  const float smin = __uint_as_float(mm[0]);
  const float smax = __uint_as_float(mm[1]);
  const float scale = scales[e0 >> 6];

  // double quantization of the scale (reference: no +smin on dequant)
  float ds;
  if (smax > smin) {
    const float ss = (smax - smin) / 255.0f;
    float q = rintf((scale - smin) / ss);          // round-half-even == jnp.round
    q = fminf(fmaxf(q, 0.0f), 255.0f);
    ds = q * ss;
  } else {
    ds = 0.0f;
  }

  // midpoints between FP4 levels, pre-scaled so we compare x directly.
  // strict '>' keeps the lower level on exact ties (== argmin first index);
  // all-false (incl. NaN) -> -3.0 == table[0], matching argmin over NaN.
  const float m0 = -2.5f   * scale, m1 = -1.75f * scale, m2 = -1.25f * scale,
              m3 = -0.875f * scale, m4 = -0.375f * scale, m5 = 0.375f * scale,
              m6 = 0.875f  * scale, m7 = 1.25f  * scale, m8 = 1.75f  * scale,
              m9 = 2.5f    * scale;

  const float4 v = *reinterpret_cast<const float4*>(x + e0);

  auto snap = [&](float xv) -> float {
    float l = -3.0f;
    l = (xv > m0) ? -2.0f  : l;
    l = (xv > m1) ? -1.5f  : l;
    l = (xv > m2) ? -1.0f  : l;
    l = (xv > m3) ? -0.75f : l;
    l = (xv > m4) ?  0.0f  : l;
    l = (xv > m5) ?  0.75f : l;
    l = (xv > m6) ?  1.0f  : l;
    l = (xv > m7) ?  1.5f  : l;
    l = (xv > m8) ?  2.0f  : l;
    l = (xv > m9) ?  3.0f  : l;
    const float xq = l * ds;
    return xv + (xq - xv);   // straight-through forward, faithful fp ordering
  };

  float4 o;
  o.x = snap(v.x);
  o.y = snap(v.y);
  o.z = snap(v.z);
  o.w = snap(v.w);
  *reinterpret_cast<float4*>(out + e0) = o;
}

extern "C" void kernel_launch(void* const* d_in, const int* in_sizes, int n_in,
                              void* d_out, int out_size, void* d_ws, size_t ws_size,
                              hipStream_t stream) {
  (void)n_in; (void)out_size; (void)ws_size;
  const float* x = (const float*)d_in[0];
  float* out = (float*)d_out;
  const long long n = (long long)in_sizes[0];      // 2048*8192 = 16,777,216
  const int nblocks = (int)(n / 64);               // 262,144 scale blocks

  unsigned* mm    = (unsigned*)d_ws;               // [0]=min bits, [1]=max bits
  float*   scales = (float*)d_ws + 4;              // 16B offset, 1 MB of scales

  fp4_init_mm<<<1, 1, 0, stream>>>(mm);

  const int grid1 = (nblocks + TPB1 - 1) / TPB1;   // 2048 WGs x 128 threads
  fp4_pass1_scales<<<grid1, TPB1, 0, stream>>>(x, scales, mm, nblocks);

  const long long nvec = n / 4;
  const int grid2 = (int)((nvec + 255) / 256);     // 16384 WGs x 256 threads
  fp4_pass2_quantdq<<<grid2, 256, 0, stream>>>(x, scales, mm, out, n);
}